// ChebConvWithSpatialAttention_3375844294973
// MI455X (gfx1250) — compile-verified
//
#include <hip/hip_runtime.h>

// ChebConv w/ spatial attention, MI455X (gfx1250), fp32 WMMA 16x16x4.
// B=8, T=16, N=1024, F_IN=32, F_OUT=64, K=3.  C = T*F_IN = 512 packed cols.
//
// Memory-bound problem (~96MB unique @ 23.3TB/s => ~4.1us floor; only 27.4GF
// total). Strategy: one WG per (b, 64-row tile), 3 k-accumulators resident in
// VGPRs so X tiles are read once per j-block; no atomics; fused ReLU.

typedef float v2f __attribute__((ext_vector_type(2)));
typedef float v8f __attribute__((ext_vector_type(8)));

__device__ __forceinline__ v8f wmma_f32(v2f a, v2f b, v8f c) {
  // D = A(16x4,f32) * B(4x16,f32) + C(16x16,f32)
  return __builtin_amdgcn_wmma_f32_16x16x4_f32(
      /*neg_a=*/false, a, /*neg_b=*/false, b,
      /*c_mod=*/(short)0, c, /*reuse_a=*/false, /*reuse_b=*/false);
}

#define B_  8
#define T_  16
#define N_  1024
#define FIN 32
#define FOUT 64
#define KC  3
#define CW  512   // packed column width = T_*FIN
#define JB  16    // j-block (K-dim tile of main GEMM)
#define XPAD 520  // 512 + 8  (bank-conflict padding; keeps half-waves disjoint)
#define APAD 20   // 16 + 4

__global__ __launch_bounds__(512)
void cheb_attn_kernel(const float* __restrict__ x,     // [8,16,1024,32]
                      const float* __restrict__ att,   // [8,1024,1024]
                      const float* __restrict__ cheb,  // [3,1024,1024]
                      const float* __restrict__ theta, // [3,32,64]
                      float* __restrict__ out)         // [8,16,1024,64]
{
  __shared__ float Xs [JB][XPAD];   // 33,280 B : X tile (j x c); reused as R tile in phase 2
  __shared__ float Att[64][APAD];   //  5,120 B : raw attention tile (i x j)
  __shared__ float Ak [64][APAD];   //  5,120 B : att*cheb tile for current k

  const int tid  = threadIdx.x;
  const int wave = tid >> 5;
  const int lane = tid & 31;
  const int b    = blockIdx.x >> 4;
  const int i0   = (blockIdx.x & 15) * 64;

  const int mh  = wave >> 3;        // row half: rows i0 + 32*mh + [0,32)
  const int c0  = (wave & 7) * 64;  // packed-column slice [c0, c0+64)
  const int lhi = lane >> 4;        // lane half (selects K=0/1 vs K=2/3 pairing)
  const int l15 = lane & 15;
  const int k0l = lhi * 2;          // per-lane K base within a 4-step

  // R accumulators: acc[k][m][n] covers rows (i0+32*mh+16m..+16) x cols (c0+16n..+16)
  v8f acc[KC][2][4];
#pragma unroll
  for (int k = 0; k < KC; ++k)
#pragma unroll
    for (int m = 0; m < 2; ++m)
#pragma unroll
      for (int n = 0; n < 4; ++n)
#pragma unroll
        for (int r = 0; r < 8; ++r) acc[k][m][n][r] = 0.0f;

  // ---------------- Phase 1: R_k = (att[b] .* cheb[k])[i-tile,:] @ X_b ----------------
  for (int jb = 0; jb < N_; jb += JB) {
    __syncthreads();  // prior reads of Xs/Att complete before overwrite

    // X tile: [JB x 512] floats, packed col c = t*32+f. 2048 float4 / 512 thr.
    for (int v = tid; v < (JB * CW) / 4; v += 512) {
      const int j  = v >> 7;            // 128 float4 per row
      const int c4 = (v & 127) << 2;
      const int t  = c4 >> 5;
      const int f  = c4 & 31;
      const float4 d = *(const float4*)(x + (((size_t)b * T_ + t) * N_ + (jb + j)) * FIN + f);
      *(float4*)(&Xs[j][c4]) = d;
    }
    // attention tile: [64 x JB]
    for (int v = tid; v < (64 * JB) / 4; v += 512) {
      const int i  = v >> 2;
      const int j4 = (v & 3) << 2;
      *(float4*)(&Att[i][j4]) =
          *(const float4*)(att + ((size_t)b * N_ + (i0 + i)) * N_ + jb + j4);
    }

#pragma unroll
    for (int k = 0; k < KC; ++k) {
      __syncthreads();  // Xs/Att ready (k=0) / previous Ak fully consumed (k>0)
      for (int v = tid; v < (64 * JB) / 4; v += 512) {
        const int i  = v >> 2;
        const int j4 = (v & 3) << 2;
        float4 d = *(const float4*)(cheb + ((size_t)k * N_ + (i0 + i)) * N_ + jb + j4);
        const float4 a = *(const float4*)(&Att[i][j4]);
        d.x *= a.x; d.y *= a.y; d.z *= a.z; d.w *= a.w;
        *(float4*)(&Ak[i][j4]) = d;
      }
      __syncthreads();

#pragma unroll
      for (int ks = 0; ks < JB / 4; ++ks) {
        const int jj = ks * 4 + k0l;  // per-lane K element base
        // B frags (4x16): elt0 -> K = jj (lanes<16: +0, lanes>=16: +2); elt1 -> jj+1
        v2f bf[4];
#pragma unroll
        for (int n = 0; n < 4; ++n) {
          const int col = c0 + n * 16 + l15;
          bf[n].x = Xs[jj][col];
          bf[n].y = Xs[jj + 1][col];
        }
#pragma unroll
        for (int m = 0; m < 2; ++m) {
          const int row = mh * 32 + m * 16 + l15;  // A: M = lane&15 in both halves
          v2f af;
          af.x = Ak[row][jj];
          af.y = Ak[row][jj + 1];
#pragma unroll
          for (int n = 0; n < 4; ++n)
            acc[k][m][n] = wmma_f32(af, bf[n], acc[k][m][n]);
        }
      }
    }
  }

  // ------------- Phase 2: out = relu( sum_k R_k @ Theta_k ), per 16-row group -------------
  // Re-layout each 16-row accumulator group through LDS (C-layout -> A-layout),
  // then WMMA against Theta_k (tiny, L2-hot, frags loaded straight from global).
#pragma unroll 1
  for (int q = 0; q < 4; ++q) {      // 16-row group: global rows i0 + 16q + [0,16)
    const int qh = q >> 1;           // owning row half
    const int qm = q & 1;            // owning m sub-tile
    v8f P[4];
#pragma unroll
    for (int n = 0; n < 4; ++n)
#pragma unroll
      for (int r = 0; r < 8; ++r) P[n][r] = 0.0f;

#pragma unroll 1
    for (int k = 0; k < KC; ++k) {
      __syncthreads();  // previous projection reads of Xs(R) done
      if (mh == qh) {
        // C layout: vgpr r -> row 16*qm + r + 8*lhi, col c0 + 16n + l15
#pragma unroll
        for (int n = 0; n < 4; ++n)
#pragma unroll
          for (int r = 0; r < 8; ++r)
            Xs[r + 8 * lhi][c0 + n * 16 + l15] = acc[k][qm][n][r];
      }
      __syncthreads();

      // wave 'w' projects timestep t = w : P += R[:, t*32..+32] @ Theta_k
      const int t = wave;
#pragma unroll
      for (int fk = 0; fk < FIN / 4; ++fk) {
        const int ff = fk * 4 + k0l;
        v2f af;
        af.x = Xs[l15][t * FIN + ff];
        af.y = Xs[l15][t * FIN + ff + 1];
#pragma unroll
        for (int n = 0; n < 4; ++n) {
          const int o = n * 16 + l15;
          v2f bf;
          bf.x = theta[((size_t)k * FIN + ff) * FOUT + o];
          bf.y = theta[((size_t)k * FIN + ff + 1) * FOUT + o];
          P[n] = wmma_f32(af, bf, P[n]);
        }
      }
    }

    // fused ReLU + store; WG exclusively owns out[b, :, i0..i0+64, :]
    const int t = wave;
#pragma unroll
    for (int n = 0; n < 4; ++n)
#pragma unroll
      for (int r = 0; r < 8; ++r) {
        float v = P[n][r];
        v = v > 0.0f ? v : 0.0f;
        const int row = i0 + q * 16 + r + 8 * lhi;
        out[(((size_t)b * T_ + t) * N_ + row) * FOUT + n * 16 + l15] = v;
      }
  }
}

extern "C" void kernel_launch(void* const* d_in, const int* in_sizes, int n_in,
                              void* d_out, int out_size, void* d_ws, size_t ws_size,
                              hipStream_t stream) {
  const float* x    = (const float*)d_in[0];  // [8,16,1024,32]
  const float* att  = (const float*)d_in[1];  // [8,1024,1024]
  const float* cheb = (const float*)d_in[2];  // [3,1024,1024]
  const float* th   = (const float*)d_in[3];  // [3,32,64]
  float* out = (float*)d_out;                 // [8,16,1024,64]

  (void)in_sizes; (void)n_in; (void)out_size; (void)d_ws; (void)ws_size;
  cheb_attn_kernel<<<dim3(B_ * (N_ / 64)), dim3(512), 0, stream>>>(x, att, cheb, th, out);
}